// RBMModel_14637248545274
// MI455X (gfx1250) — compile-verified
//
#include <hip/hip_runtime.h>
#include <hip/hip_bf16.h>

// Problem constants (match reference)
constexpr int BATCH = 256;
constexpr int TCAND = 64;
constexpr int DDIM  = 2048;

typedef float v2f __attribute__((ext_vector_type(2)));
typedef float v8f __attribute__((ext_vector_type(8)));

// ---------------------------------------------------------------------------
// Kernel 1: transformed = experimental @ W   (fp32 WMMA 16x16x4)
//   A = experimental [256, 2048] row-major
//   Wm = W           [2048, 2048] row-major (k fastest over rows)
//   C = transformed  [256, 2048]
// Block: 256 threads = 8 waves arranged 2(M) x 4(N); block tile 32x64, KC=32.
// ---------------------------------------------------------------------------
constexpr int TM = 32;
constexpr int TN = 64;
constexpr int KC = 32;

__global__ __launch_bounds__(256)
void rbm_gemm_wmma_f32(const float* __restrict__ A,
                       const float* __restrict__ Wm,
                       float* __restrict__ C) {
  __shared__ float As[TM][KC + 1];   // +1 pad: conflict-free column reads
  __shared__ float Bs[KC][TN];       // rows read broadcast/consecutive: no pad needed

  const int tid = threadIdx.x;
  const int wv  = tid >> 5;          // wave 0..7
  const int ln  = tid & 31;          // lane 0..31
  const int lm  = ln & 15;           // lane mod 16
  const int hi  = ln >> 4;           // half-wave select (K pairing per ISA layout)

  const int waveM = wv >> 2;         // 0..1 -> 16-row slice of block tile
  const int waveN = wv & 3;          // 0..3 -> 16-col slice of block tile
  const int m_off = waveM * 16;
  const int n_off = waveN * 16;

  const int m0 = blockIdx.y * TM;    // block row origin in C
  const int n0 = blockIdx.x * TN;    // block col origin in C

  v8f acc = {};                      // 16x16 f32 accumulator (8 VGPRs)

  for (int k0 = 0; k0 < DDIM; k0 += KC) {
    // --- stage A tile: 32 rows x 32 cols = 256 float4 loads, 1 per thread ---
    {
      const int r  = tid >> 3;             // 0..31
      const int c4 = (tid & 7) << 2;       // 0,4,...,28
      const float4 av = *(const float4*)(A + (size_t)(m0 + r) * DDIM + k0 + c4);
      As[r][c4 + 0] = av.x;
      As[r][c4 + 1] = av.y;
      As[r][c4 + 2] = av.z;
      As[r][c4 + 3] = av.w;
    }
    // --- stage B tile: 32 rows x 64 cols = 512 float4 loads, 2 per thread ---
    #pragma unroll
    for (int i = 0; i < 2; ++i) {
      const int idx = tid + i * 256;       // 0..511
      const int r   = idx >> 4;            // 0..31
      const int c4  = (idx & 15) << 2;     // 0..60
      *(float4*)&Bs[r][c4] =
          *(const float4*)(Wm + (size_t)(k0 + r) * DDIM + n0 + c4);
    }
    __syncthreads();

    // --- 8 WMMA steps over the K-chunk ---
    #pragma unroll
    for (int kk = 0; kk < KC; kk += 4) {
      // A fragment (16x4 f32): lane lm = row M; half-wave picks K pair {0,1}/{2,3}
      v2f a, b;
      a.x = As[m_off + lm][kk + hi * 2 + 0];
      a.y = As[m_off + lm][kk + hi * 2 + 1];
      // B fragment (4x16 f32): lane lm = col N; half-wave picks K pair
      b.x = Bs[kk + hi * 2 + 0][n_off + lm];
      b.y = Bs[kk + hi * 2 + 1][n_off + lm];
      acc = __builtin_amdgcn_wmma_f32_16x16x4_f32(
          /*neg_a=*/false, a, /*neg_b=*/false, b,
          /*c_mod=*/(short)0, acc, /*reuse_a=*/false, /*reuse_b=*/false);
    }
    __syncthreads();
  }

  // --- write 16x16 tile per ISA C/D layout: VGPR i -> row (i + hi*8), col lm ---
  #pragma unroll
  for (int i = 0; i < 8; ++i) {
    C[(size_t)(m0 + m_off + hi * 8 + i) * DDIM + n0 + n_off + lm] = acc[i];
  }
}

// ---------------------------------------------------------------------------
// Kernel 2: per-spectrum scores. One block (256 thr / 8 waves) per b.
//   u[d] = transformed[b][d] + hb[d]            (LDS, 8 KB)
//   out[b][0]    = dot(tr,cand) + dot(exp,vb) + dot(cand,hb)
//   out[b][1+t]  = dot(theo[b][t], u) + dot(exp,vb)
// Memory-bound on streaming theoretical (134 MB total): float4 coalesced.
// ---------------------------------------------------------------------------
__global__ __launch_bounds__(256)
void rbm_score(const float* __restrict__ experimental,
               const float* __restrict__ theoretical,
               const float* __restrict__ candidate,
               const float* __restrict__ vb,
               const float* __restrict__ hb,
               const float* __restrict__ transformed,
               float* __restrict__ out) {
  const int b   = blockIdx.x;
  const int tid = threadIdx.x;
  const int wv  = tid >> 5;
  const int ln  = tid & 31;

  __shared__ __align__(16) float u[DDIM];
  __shared__ float red[3][8];

  const float* e  = experimental + (size_t)b * DDIM;
  const float* c  = candidate    + (size_t)b * DDIM;
  const float* tr = transformed  + (size_t)b * DDIM;

  float vb_p = 0.f, cb_p = 0.f, pos_p = 0.f;
  for (int d = tid; d < DDIM; d += 256) {
    const float td  = tr[d];
    const float cd  = c[d];
    const float hbd = hb[d];
    vb_p  += e[d] * vb[d];
    cb_p  += cd * hbd;
    pos_p += td * cd;
    u[d] = td + hbd;
  }
  #pragma unroll
  for (int off = 16; off; off >>= 1) {
    vb_p  += __shfl_xor(vb_p,  off, 32);
    cb_p  += __shfl_xor(cb_p,  off, 32);
    pos_p += __shfl_xor(pos_p, off, 32);
  }
  if (ln == 0) { red[0][wv] = vb_p; red[1][wv] = cb_p; red[2][wv] = pos_p; }
  __syncthreads();

  float vbias = 0.f, cbias = 0.f, pos = 0.f;
  #pragma unroll
  for (int i = 0; i < 8; ++i) {
    vbias += red[0][i]; cbias += red[1][i]; pos += red[2][i];
  }
  if (tid == 0) out[(size_t)b * (TCAND + 1)] = pos + vbias + cbias;

  // negative phase: wave wv owns rows t = wv, wv+8, ... (8 rows each)
  for (int t = wv; t < TCAND; t += 8) {
    const float* th = theoretical + ((size_t)b * TCAND + t) * DDIM;
    float s = 0.f;
    for (int d = ln * 4; d < DDIM; d += 32 * 4) {
      const float4 tv = *(const float4*)(th + d);
      const float4 uv = *(const float4*)&u[d];
      s += tv.x * uv.x + tv.y * uv.y + tv.z * uv.z + tv.w * uv.w;
    }
    #pragma unroll
    for (int off = 16; off; off >>= 1) s += __shfl_xor(s, off, 32);
    if (ln == 0) out[(size_t)b * (TCAND + 1) + 1 + t] = s + vbias;
  }
}

// ---------------------------------------------------------------------------
extern "C" void kernel_launch(void* const* d_in, const int* in_sizes, int n_in,
                              void* d_out, int out_size, void* d_ws, size_t ws_size,
                              hipStream_t stream) {
  const float* experimental = (const float*)d_in[0];  // [256, 2048]
  const float* theoretical  = (const float*)d_in[1];  // [256, 64, 2048]
  const float* candidate    = (const float*)d_in[2];  // [256, 2048]
  const float* Wm           = (const float*)d_in[3];  // [2048, 2048]
  const float* vb           = (const float*)d_in[4];  // [2048]
  const float* hb           = (const float*)d_in[5];  // [2048]
  float* out         = (float*)d_out;                 // [256, 65]
  float* transformed = (float*)d_ws;                  // [256, 2048] scratch (2 MB)

  dim3 g1(DDIM / TN, BATCH / TM);   // (32, 8) = 256 workgroups
  rbm_gemm_wmma_f32<<<g1, 256, 0, stream>>>(experimental, Wm, transformed);

  rbm_score<<<BATCH, 256, 0, stream>>>(experimental, theoretical, candidate,
                                       vb, hb, transformed, out);
}